// MLFF_dmirror_3659312136734
// MI455X (gfx1250) — compile-verified
//
#include <hip/hip_runtime.h>

typedef float v2f __attribute__((ext_vector_type(2)));
typedef float v8f __attribute__((ext_vector_type(8)));

// D = A(16x4) * B(4x16) + C, f32, wave32.
#define WMMA_F32(a, b, c) \
    __builtin_amdgcn_wmma_f32_16x16x4_f32(false, (a), false, (b), (short)0, (c), false, false)

constexpr int BATCH  = 8;
constexpr int NATOMS = 256;
constexpr int DF     = 42;    // features
constexpr int H1     = 64;
constexpr int H2     = 32;
constexpr int ROWS   = BATCH * NATOMS;   // 2048 (= 128 tiles of 16)
constexpr int JF     = NATOMS * DF;      // 10752

__device__ __forceinline__ float softplus_f(float z) {
    return fmaxf(z, 0.0f) + log1pf(expf(-fabsf(z)));
}
__device__ __forceinline__ float sigmoid_f(float z) {
    return 1.0f / (1.0f + expf(-z));
}

// One wave (32 lanes) per 16-row tile. Computes Ei[row] and dEdFeat[row][0..41].
__global__ __launch_bounds__(32)
void mlp_wmma_kernel(const float* __restrict__ image,
                     const float* __restrict__ W1, const float* __restrict__ b1,
                     const float* __restrict__ W2, const float* __restrict__ b2,
                     const float* __restrict__ W3, const float* __restrict__ b3,
                     float* __restrict__ gws,   // [ROWS][DF]
                     float* __restrict__ eiws)  // [ROWS]
{
    __shared__ float sW1[48 * 64];   // W1 padded to 48 rows (rows 42..47 zero)
    __shared__ float sW2[64 * 32];
    __shared__ float sW3[32];
    __shared__ float sb1[64];
    __shared__ float sb2[32];
    __shared__ float sX [16 * 44];   // image tile, K padded to 44 (cols 42,43 zero)
    __shared__ float sA1[16 * 64];
    __shared__ float sA2[16 * 32];
    __shared__ float sG2[16 * 32];
    __shared__ float sG1[16 * 64];

    const int lane = threadIdx.x;       // 0..31
    const int half = lane >> 4;         // 0 or 1
    const int l    = lane & 15;         // m (for A) / n (for B,C,D)
    const int tile = blockIdx.x;        // 0..127
    const int row0 = tile * 16;

    // --- stage weights / inputs in LDS ---
    for (int idx = lane; idx < 48 * 64; idx += 32) {
        int r = idx >> 6, c = idx & 63;
        sW1[idx] = (r < 42) ? W1[r * 64 + c] : 0.0f;
    }
    for (int idx = lane; idx < 64 * 32; idx += 32) sW2[idx] = W2[idx];
    sW3[lane] = W3[lane];
    for (int idx = lane; idx < 64; idx += 32) sb1[idx] = b1[idx];
    sb2[lane] = b2[lane];
    for (int idx = lane; idx < 16 * 44; idx += 32) {
        int r = idx / 44, c = idx % 44;
        sX[idx] = (c < 42) ? image[(row0 + r) * DF + c] : 0.0f;
    }
    __syncthreads();

    // ---------------- GEMM1: z1[16x64] = X[16x44] @ W1[44x64] + b1 ----------------
    v8f z1t[4];
    #pragma unroll
    for (int t = 0; t < 4; ++t) {
        float bv = sb1[t * 16 + l];
        v8f c;
        #pragma unroll
        for (int i = 0; i < 8; ++i) c[i] = bv;
        #pragma unroll
        for (int k0 = 0; k0 < 44; k0 += 4) {
            v2f a, b;
            a.x = sX[l * 44 + k0 + 2 * half];
            a.y = sX[l * 44 + k0 + 2 * half + 1];
            b.x = sW1[(k0 + 2 * half)     * 64 + t * 16 + l];
            b.y = sW1[(k0 + 2 * half + 1) * 64 + t * 16 + l];
            c = WMMA_F32(a, b, c);
        }
        z1t[t] = c;
    }

    // a1 = softplus(z1) -> LDS ; keep sigmoid(z1) in regs
    v8f sig1[4];
    #pragma unroll
    for (int t = 0; t < 4; ++t) {
        #pragma unroll
        for (int i = 0; i < 8; ++i) {
            float z = z1t[t][i];
            sA1[(i + 8 * half) * 64 + t * 16 + l] = softplus_f(z);
            sig1[t][i] = sigmoid_f(z);
        }
    }
    __syncthreads();

    // ---------------- GEMM2: z2[16x32] = a1[16x64] @ W2[64x32] + b2 ----------------
    v8f z2t[2];
    #pragma unroll
    for (int t = 0; t < 2; ++t) {
        float bv = sb2[t * 16 + l];
        v8f c;
        #pragma unroll
        for (int i = 0; i < 8; ++i) c[i] = bv;
        #pragma unroll
        for (int k0 = 0; k0 < 64; k0 += 4) {
            v2f a, b;
            a.x = sA1[l * 64 + k0 + 2 * half];
            a.y = sA1[l * 64 + k0 + 2 * half + 1];
            b.x = sW2[(k0 + 2 * half)     * 32 + t * 16 + l];
            b.y = sW2[(k0 + 2 * half + 1) * 32 + t * 16 + l];
            c = WMMA_F32(a, b, c);
        }
        z2t[t] = c;
    }

    // a2 = softplus(z2) -> LDS ; g2 = sigmoid(z2)*W3[n] -> LDS
    #pragma unroll
    for (int t = 0; t < 2; ++t) {
        float w3v = sW3[t * 16 + l];
        #pragma unroll
        for (int i = 0; i < 8; ++i) {
            float z = z2t[t][i];
            sA2[(i + 8 * half) * 32 + t * 16 + l] = softplus_f(z);
            sG2[(i + 8 * half) * 32 + t * 16 + l] = sigmoid_f(z) * w3v;
        }
    }
    __syncthreads();

    // ---------------- Ei: [16x16] = a2[16x32] @ W3col[32x16] (only col 0 real) ----
    v8f ce;
    #pragma unroll
    for (int i = 0; i < 8; ++i) ce[i] = 0.0f;
    #pragma unroll
    for (int k0 = 0; k0 < 32; k0 += 4) {
        v2f a, b;
        a.x = sA2[l * 32 + k0 + 2 * half];
        a.y = sA2[l * 32 + k0 + 2 * half + 1];
        b.x = (l == 0) ? sW3[k0 + 2 * half]     : 0.0f;
        b.y = (l == 0) ? sW3[k0 + 2 * half + 1] : 0.0f;
        ce = WMMA_F32(a, b, ce);
    }
    {
        float b3v = b3[0];
        if (l == 0) {   // lanes 0 (m=v) and 16 (m=v+8) hold column 0
            #pragma unroll
            for (int v = 0; v < 8; ++v)
                eiws[row0 + 8 * half + v] = ce[v] + b3v;
        }
    }

    // ---------------- GEMM3: h1[16x64] = g2[16x32] @ W2^T[32x64] ----------------
    #pragma unroll
    for (int t = 0; t < 4; ++t) {
        v8f c;
        #pragma unroll
        for (int i = 0; i < 8; ++i) c[i] = 0.0f;
        #pragma unroll
        for (int k0 = 0; k0 < 32; k0 += 4) {
            v2f a, b;
            a.x = sG2[l * 32 + k0 + 2 * half];
            a.y = sG2[l * 32 + k0 + 2 * half + 1];
            b.x = sW2[(t * 16 + l) * 32 + k0 + 2 * half];      // W2^T[k][n] = W2[n][k]
            b.y = sW2[(t * 16 + l) * 32 + k0 + 2 * half + 1];
            c = WMMA_F32(a, b, c);
        }
        // g1 = sigmoid(z1) * h1  -> LDS (same D layout as z1)
        #pragma unroll
        for (int i = 0; i < 8; ++i)
            sG1[(i + 8 * half) * 64 + t * 16 + l] = sig1[t][i] * c[i];
    }
    __syncthreads();

    // ---------------- GEMM4: dEdF[16x48] = g1[16x64] @ W1^T[64x48] ----------------
    #pragma unroll
    for (int t = 0; t < 3; ++t) {
        v8f c;
        #pragma unroll
        for (int i = 0; i < 8; ++i) c[i] = 0.0f;
        #pragma unroll
        for (int k0 = 0; k0 < 64; k0 += 4) {
            v2f a, b;
            a.x = sG1[l * 64 + k0 + 2 * half];
            a.y = sG1[l * 64 + k0 + 2 * half + 1];
            b.x = sW1[(t * 16 + l) * 64 + k0 + 2 * half];      // W1^T[k][d] = W1[d][k]
            b.y = sW1[(t * 16 + l) * 64 + k0 + 2 * half + 1];
            c = WMMA_F32(a, b, c);
        }
        int d = t * 16 + l;
        if (d < DF) {
            #pragma unroll
            for (int v = 0; v < 8; ++v)
                gws[(row0 + v + 8 * half) * DF + d] = c[v];
        }
    }
}

// Streaming contraction: Force[b,i,c] = sum_jf g[b,jf] * dfeat[b,i,jf,c].
// 256 blocks x 256 threads; block handles (b, 8 atoms); one wave per atom row.
__global__ __launch_bounds__(256)
void force_kernel(const float* __restrict__ dfeat,
                  const float* __restrict__ gws,
                  float* __restrict__ out)
{
    __shared__ __align__(16) float gsh[JF];   // 43 KB, fits WGP LDS easily

    const int tid = threadIdx.x;
    const int blk = blockIdx.x;        // 0..255
    const int b   = blk >> 5;          // 32 blocks per batch
    const int i0  = (blk & 31) * 8;

    for (int idx = tid; idx < JF; idx += 256) gsh[idx] = gws[b * JF + idx];
    __syncthreads();

    const int wave = tid >> 5;
    const int lane = tid & 31;
    const int i    = i0 + wave;

    const float* row = dfeat + (size_t)(b * NATOMS + i) * (size_t)(JF * 3);
    const float4* g4 = (const float4*)gsh;

    float ax = 0.0f, ay = 0.0f, az = 0.0f;
    // 10752 jf = 84 iters * 32 lanes * 4 jf ; lane handles jf=4q..4q+3 (12 floats).
    for (int it = 0; it < 84; ++it) {
        int q = it * 32 + lane;                         // float4-group index of g
        const float4* p = (const float4*)(row + (size_t)q * 12);  // 48B aligned
        float4 d0 = p[0], d1 = p[1], d2 = p[2];
        float4 gv = g4[q];
        ax = fmaf(d0.x, gv.x, fmaf(d0.w, gv.y, fmaf(d1.z, gv.z, fmaf(d2.y, gv.w, ax))));
        ay = fmaf(d0.y, gv.x, fmaf(d1.x, gv.y, fmaf(d1.w, gv.z, fmaf(d2.z, gv.w, ay))));
        az = fmaf(d0.z, gv.x, fmaf(d1.y, gv.y, fmaf(d2.x, gv.z, fmaf(d2.w, gv.w, az))));
    }
    #pragma unroll
    for (int off = 16; off; off >>= 1) {
        ax += __shfl_xor(ax, off, 32);
        ay += __shfl_xor(ay, off, 32);
        az += __shfl_xor(az, off, 32);
    }
    if (lane == 0) {
        float* f = out + 8 + (size_t)(b * NATOMS + i) * 3;
        f[0] = ax; f[1] = ay; f[2] = az;
    }
}

// Etot[b] = sum_n Ei[b,n]; one wave per batch element.
__global__ __launch_bounds__(256)
void etot_kernel(const float* __restrict__ eiws, float* __restrict__ out)
{
    const int b = threadIdx.x >> 5;
    const int lane = threadIdx.x & 31;
    float s = 0.0f;
    #pragma unroll
    for (int k = 0; k < 8; ++k) s += eiws[b * NATOMS + lane + 32 * k];
    #pragma unroll
    for (int off = 16; off; off >>= 1) s += __shfl_xor(s, off, 32);
    if (lane == 0) out[b] = s;
}

extern "C" void kernel_launch(void* const* d_in, const int* in_sizes, int n_in,
                              void* d_out, int out_size, void* d_ws, size_t ws_size,
                              hipStream_t stream)
{
    (void)in_sizes; (void)n_in; (void)out_size; (void)ws_size;

    const float* image = (const float*)d_in[0];
    const float* dfeat = (const float*)d_in[1];
    const float* W1    = (const float*)d_in[2];
    const float* b1    = (const float*)d_in[3];
    const float* W2    = (const float*)d_in[4];
    const float* b2    = (const float*)d_in[5];
    const float* W3    = (const float*)d_in[6];
    const float* b3    = (const float*)d_in[7];

    float* out  = (float*)d_out;            // [0..7]=Etot, [8..]=Force[B,N,3]
    float* gws  = (float*)d_ws;             // dEdFeat: ROWS*DF floats
    float* eiws = gws + (size_t)ROWS * DF;  // Ei: ROWS floats

    mlp_wmma_kernel<<<ROWS / 16, 32, 0, stream>>>(image, W1, b1, W2, b2, W3, b3, gws, eiws);
    force_kernel<<<256, 256, 0, stream>>>(dfeat, gws, out);
    etot_kernel<<<1, 256, 0, stream>>>(eiws, out);
}